// Butterfly_18511309045994
// MI455X (gfx1250) — compile-verified
//
#include <hip/hip_runtime.h>

// Butterfly multiply (B=8192, N=4096, log_n=12, nstacks=nblocks=1, increasing stride)
// Strategy: compose each group of 4 butterfly levels (which only mix one hex digit
// of the element index) into 3 x 256 dense 16x16 matrices (shared across the whole
// batch), then apply them as batched 16x16x16 GEMMs with V_WMMA_F32_16X16X4_F32,
// fully fused in LDS so x makes exactly one HBM round trip (~256MB -> ~11.5us roofline).

typedef float v2f __attribute__((ext_vector_type(2)));
typedef float v8f __attribute__((ext_vector_type(8)));

#define LOG_N       12
#define NELEM       4096
#define BATCH       8192
#define ROWS_TILE   16            // WMMA N dimension = rows per workgroup tile
#define LDS_STRIDE  4100          // 4096 + 4 floats pad: row stride % 64 banks == 4

// Position of digit value t within a row for stage g, context c.
//  g=0: digit = bits 0-3, context = bits 4-11
//  g=1: digit = bits 4-7, context = {bits 8-11, bits 0-3}
//  g=2: digit = bits 8-11, context = bits 0-7
__device__ __forceinline__ void ctx_pos(int g, int c, int& base, int& step) {
    if (g == 0)      { base = c << 4;                        step = 1;   }
    else if (g == 1) { base = ((c & 0xF0) << 4) | (c & 15);  step = 16;  }
    else             { base = c;                             step = 256; }
}

// ---------------------------------------------------------------------------
// Kernel 1: compose 4 butterfly levels per stage into 16x16 matrices.
// One thread per (stage g, context c, column k): 3*256*16 = 12288 threads.
// M is row-major: M[((g*256 + c)*16 + i)*16 + k]
// ---------------------------------------------------------------------------
__global__ void butterfly_compose(const float* __restrict__ tw,
                                  float* __restrict__ M) {
    int tid = blockIdx.x * blockDim.x + threadIdx.x;
    if (tid >= 3 * 256 * 16) return;
    int k = tid & 15;
    int c = (tid >> 4) & 255;
    int g = tid >> 12;

    int base, step;
    ctx_pos(g, c, base, step);

    float v[16];
#pragma unroll
    for (int t = 0; t < 16; ++t) v[t] = (t == k) ? 1.0f : 0.0f;

#pragma unroll
    for (int j = 0; j < 4; ++j) {              // level within stage
        const int idx = 4 * g + j;             // global level index
        const int S   = 1 << idx;              // global stride
#pragma unroll
        for (int u = 0; u < 8; ++u) {          // 8 pairs in the 16-digit space
            const int off = u & ((1 << j) - 1);
            const int blk = u >> j;
            const int t0  = (blk << (j + 1)) | off;
            const int t1  = t0 + (1 << j);
            const int p0  = base + t0 * step;  // full index of pair's first elem
            // twiddle index = p0 with bit 'idx' dropped
            const int w   = ((p0 >> (idx + 1)) << idx) | (p0 & (S - 1));
            const float* q = tw + idx * (2048 * 4) + w * 4;  // [i][k] 2x2
            const float a00 = q[0], a01 = q[1], a10 = q[2], a11 = q[3];
            const float x0 = v[t0], x1 = v[t1];
            v[t0] = a00 * x0 + a01 * x1;
            v[t1] = a10 * x0 + a11 * x1;
        }
    }

    float* out = M + ((g * 256 + c) * 16) * 16 + k;   // column k
#pragma unroll
    for (int i = 0; i < 16; ++i) out[i * 16] = v[i];
}

// ---------------------------------------------------------------------------
// Kernel 2: apply the 3 stages to a 16-row tile held in LDS, via WMMA f32.
//   D(16 digits x 16 rows) = A(16x16 matrix, K-chunked) x B(digits x rows)
// 512 blocks x 256 threads (8 wave32); each wave owns 32 contexts per stage.
// ---------------------------------------------------------------------------
__global__ void __launch_bounds__(256, 1)
butterfly_apply(const float* __restrict__ x,
                const float* __restrict__ M,
                float* __restrict__ out) {
    extern __shared__ float X[];               // [16][LDS_STRIDE]

    const int r0 = blockIdx.x * ROWS_TILE;

    // ---- cooperative coalesced load: 16 rows x 4096 floats -> LDS ----
    {
        const float4* src = (const float4*)(x + (size_t)r0 * NELEM);
        for (int i = threadIdx.x; i < ROWS_TILE * (NELEM / 4); i += 256) {
            const int row = i >> 10;           // 1024 float4 per row
            const int c4  = i & 1023;
            float4 val = src[row * 1024 + c4];
            *(float4*)&X[row * LDS_STRIDE + c4 * 4] = val;
        }
    }

    const int lane = threadIdx.x & 31;
    const int wave = threadIdx.x >> 5;
    const int r    = lane & 15;                // B/D column (batch row), A row (M)
    const int hi   = lane >> 4;                // lane-half selector

    for (int g = 0; g < 3; ++g) {
        __syncthreads();
        for (int ci = 0; ci < 32; ++ci) {
            const int c = (wave << 5) | ci;    // context 0..255
            int base, step;
            ctx_pos(g, c, base, step);

            const float* Mrow = M + ((g * 256 + c) * 16 + r) * 16;  // A row m=r
            const float* Brow = X + r * LDS_STRIDE + base;          // B col n=r

            v8f acc = {};
#pragma unroll
            for (int q = 0; q < 4; ++q) {      // K = 16 in chunks of 4
                const int kb = 4 * q + 2 * hi;
                // A 16x4 f32 layout: lanes 0-15 m=lane; V0:K={kb}, V1:K={kb+1}
                //                    lanes16-31 same m; V0:K={kb(+2)}, V1:K={kb+3}
                v2f a;
                a.x = Mrow[kb];
                a.y = Mrow[kb + 1];
                // B 4x16 f32 layout: lanes 0-15 hold K first half (V0,V1 ascending),
                //                    lanes16-31 hold K second half; N = lane&15
                v2f b;
                b.x = Brow[kb * step];
                b.y = Brow[(kb + 1) * step];
                acc = __builtin_amdgcn_wmma_f32_16x16x4_f32(
                    /*neg_a=*/false, a, /*neg_b=*/false, b,
                    /*c_mod=*/(short)0, acc, /*reuse_a=*/false, /*reuse_b=*/false);
            }

            // D layout: VGPR j -> M = j + 8*hi, N = lane&15. Scatter in place
            // (contexts touch disjoint positions within a stage).
            float* Drow = X + r * LDS_STRIDE + base;
#pragma unroll
            for (int j = 0; j < 8; ++j) {
                const int i = j + 8 * hi;
                Drow[i * step] = acc[j];
            }
        }
    }

    __syncthreads();

    // ---- cooperative coalesced store: LDS -> d_out ----
    {
        float4* dst = (float4*)(out + (size_t)r0 * NELEM);
        for (int i = threadIdx.x; i < ROWS_TILE * (NELEM / 4); i += 256) {
            const int row = i >> 10;
            const int c4  = i & 1023;
            dst[row * 1024 + c4] = *(const float4*)&X[row * LDS_STRIDE + c4 * 4];
        }
    }
}

// ---------------------------------------------------------------------------
extern "C" void kernel_launch(void* const* d_in, const int* in_sizes, int n_in,
                              void* d_out, int out_size, void* d_ws, size_t ws_size,
                              hipStream_t stream) {
    (void)in_sizes; (void)n_in; (void)out_size; (void)ws_size;

    const float* x  = (const float*)d_in[0];   // (8192, 4096) f32
    const float* tw = (const float*)d_in[1];   // (1,1,12,2048,2,2) f32
    float*       o  = (float*)d_out;
    float*       M  = (float*)d_ws;            // needs 3*256*256*4 = 768 KB

    // Stage-matrix precompute: 12288 threads.
    butterfly_compose<<<48, 256, 0, stream>>>(tw, M);

    const size_t lds_bytes = (size_t)ROWS_TILE * LDS_STRIDE * sizeof(float); // 256.25 KB
    (void)hipFuncSetAttribute((const void*)butterfly_apply,
                              hipFuncAttributeMaxDynamicSharedMemorySize,
                              (int)lds_bytes);

    butterfly_apply<<<BATCH / ROWS_TILE, 256, lds_bytes, stream>>>(x, M, o);
}